// AttentionNorm_38439957299912
// MI455X (gfx1250) — compile-verified
//
#include <hip/hip_runtime.h>
#include <hip/hip_bf16.h>

// GAT edge-softmax for MI455X (gfx1250, wave32).
//
//  K1: init segmax/segsum tables
//  K2: node scores S = X * W via V_WMMA_F32_16X16X4_F32 (fp32-exact GEMM)
//  K3: per-edge alpha + leaky-relu, store to d_out, atomic segment-max (u32-encoded)
//  K4: per-edge exp(alpha - max[col]), store to d_out, atomic segment-sum (f32 add)
//  K5: per-edge normalize ex / (sum[col] + eps)
//
// Memory-bound: ~0.65 GB total traffic, intermediates (102 MB) fit in 192 MB L2.

typedef __attribute__((ext_vector_type(2))) float v2f;
typedef __attribute__((ext_vector_type(8))) float v8f;

#define NEG_SLOPE 0.2f
#define EPS_F 1e-16f

// Monotone float<->uint order-preserving encoding for atomicMax on u32.
__device__ __forceinline__ unsigned enc_f32(float x) {
  unsigned b = __float_as_uint(x);
  return (b & 0x80000000u) ? ~b : (b | 0x80000000u);
}
__device__ __forceinline__ float dec_f32(unsigned e) {
  unsigned b = (e & 0x80000000u) ? (e & 0x7FFFFFFFu) : ~e;
  return __uint_as_float(b);
}
__device__ __forceinline__ float lrelu(float x) {
  return x >= 0.0f ? x : NEG_SLOPE * x;
}

// ---------------- K1: zero the segment tables -------------------------------
__global__ void init_tables(unsigned* __restrict__ segmax,
                            float* __restrict__ segsum, int n4) {
  int i = blockIdx.x * blockDim.x + threadIdx.x;
  if (i < n4) {
    segmax[i] = 0u;     // enc-space identity below any real float
    segsum[i] = 0.0f;
  }
}

// ---------------- K2: node scores via WMMA ----------------------------------
// One wave computes a 16-node tile: D[16x16] = A[16x128] * W[128x16], where
// W col n<4  = att_src head n (non-zero only for K in [n*32, n*32+32))
// W col 4..7 = att_dst head n-4 (same band structure); cols 8..15 = 0.
// Result: sc[node][n] with n=0..3 -> s_src[h], n=4..7 -> s_dst[h].
//
// B fragments are preloaded UNCONDITIONALLY (branch-free) and selected with
// v_cndmask in the unrolled K-loop — avoids exec-masked scalar-load blocks.
__global__ void node_scores_wmma(const float* __restrict__ X,
                                 const float* __restrict__ att_src,
                                 const float* __restrict__ att_dst,
                                 float* __restrict__ sc, int N) {
  const int lane  = threadIdx.x & 31;
  const int wave  = threadIdx.x >> 5;
  const int tile  = blockIdx.x * (blockDim.x >> 5) + wave;
  const int node0 = tile * 16;
  if (node0 >= N) return;                 // whole-wave exit: EXEC stays all-1s for WMMA

  const int m  = lane & 15;               // A row index; also B/D column index
  const int hi = lane >> 4;               // K-pair selector (K offset 0 or 2)

  // B column setup for column n = m
  const bool bvalid = (m < 8);
  const int  head   = m & 3;
  const float* attp = ((m < 4) ? att_src : att_dst) + head * 32;

  // Preload the 8 B-pairs this lane can ever need: attp[4j + 2*hi .. +1].
  // Offsets are always < 32, so loads are unconditional (in-bounds even for
  // lanes whose column is structurally zero — values just go unused).
  v2f bp[8];
  #pragma unroll
  for (int j = 0; j < 8; ++j)
    bp[j] = *(const v2f*)(attp + j * 4 + hi * 2);

  // A row pointer (clamped for generality; N here is a multiple of 16)
  const int nidx = (node0 + m) < N ? (node0 + m) : (N - 1);
  const float* arow = X + (size_t)nidx * 128;

  const v2f bzero = {0.0f, 0.0f};
  v8f acc = {};
  #pragma unroll
  for (int kk = 0; kk < 32; ++kk) {
    // A fragment per ISA 16x4 f32 layout: v0 = K=c0, v1 = K=c0+1
    v2f a = *(const v2f*)(arow + kk * 4 + hi * 2);
    // B fragment: nonzero only inside this head's K-band (8 of 32 steps)
    const bool act = bvalid && ((kk >> 3) == head);
    v2f b = act ? bp[kk & 7] : bzero;     // v_cndmask pair, branch-free
    acc = __builtin_amdgcn_wmma_f32_16x16x4_f32(
        /*neg_a=*/false, a, /*neg_b=*/false, b,
        /*c_mod=*/(short)0, acc, /*reuse_a=*/false, /*reuse_b=*/false);
  }

  // D layout: lanes 0-15 hold M=r (VGPR r), N=lane; lanes 16-31 hold M=8+r
  if (m < 8) {
    #pragma unroll
    for (int r = 0; r < 8; ++r) {
      const int node = node0 + r + 8 * hi;
      if (node < N) sc[(size_t)node * 8 + m] = acc[r];
    }
  }
}

// ---------------- K3: alpha + leaky-relu + segment max ----------------------
__global__ void edge_alpha_max(const float* __restrict__ sc,
                               const float* __restrict__ ew,
                               const float* __restrict__ att_edge,
                               const int* __restrict__ row,
                               const int* __restrict__ col,
                               float* __restrict__ alpha_out,
                               unsigned* __restrict__ segmax, int E) {
  int e = blockIdx.x * blockDim.x + threadIdx.x;
  if (e >= E) return;
  const int   r = row[e];
  const int   c = col[e];
  const float w = ew[e];
  const float4 ss = *(const float4*)(sc + (size_t)r * 8);      // s_src[r][0..3]
  const float4 sd = *(const float4*)(sc + (size_t)c * 8 + 4);  // s_dst[c][0..3]
  const float4 ae = *(const float4*)att_edge;                  // uniform
  float4 a;
  a.x = lrelu(ss.x + sd.x + ae.x * w);
  a.y = lrelu(ss.y + sd.y + ae.y * w);
  a.z = lrelu(ss.z + sd.z + ae.z * w);
  a.w = lrelu(ss.w + sd.w + ae.w * w);
  *(float4*)(alpha_out + (size_t)e * 4) = a;
  unsigned* sm = segmax + (size_t)c * 4;
  atomicMax(sm + 0, enc_f32(a.x));
  atomicMax(sm + 1, enc_f32(a.y));
  atomicMax(sm + 2, enc_f32(a.z));
  atomicMax(sm + 3, enc_f32(a.w));
}

// ---------------- K4: exp(alpha - max) + segment sum ------------------------
__global__ void edge_exp_sum(const int* __restrict__ col,
                             const unsigned* __restrict__ segmax,
                             float* __restrict__ ex_inout,
                             float* __restrict__ segsum, int E) {
  int e = blockIdx.x * blockDim.x + threadIdx.x;
  if (e >= E) return;
  const int c = col[e];
  float4 a = *(const float4*)(ex_inout + (size_t)e * 4);
  uint4  m = *(const uint4*)(segmax + (size_t)c * 4);
  float4 ex;
  ex.x = __expf(a.x - dec_f32(m.x));
  ex.y = __expf(a.y - dec_f32(m.y));
  ex.z = __expf(a.z - dec_f32(m.z));
  ex.w = __expf(a.w - dec_f32(m.w));
  *(float4*)(ex_inout + (size_t)e * 4) = ex;
  float* sp = segsum + (size_t)c * 4;
  atomicAdd(sp + 0, ex.x);
  atomicAdd(sp + 1, ex.y);
  atomicAdd(sp + 2, ex.z);
  atomicAdd(sp + 3, ex.w);
}

// ---------------- K5: normalize ---------------------------------------------
__global__ void edge_norm(const int* __restrict__ col,
                          const float* __restrict__ segsum,
                          float* __restrict__ out, int E) {
  int e = blockIdx.x * blockDim.x + threadIdx.x;
  if (e >= E) return;
  const int c = col[e];
  float4 s  = *(const float4*)(segsum + (size_t)c * 4);
  float4 ex = *(const float4*)(out + (size_t)e * 4);
  ex.x = ex.x / (s.x + EPS_F);
  ex.y = ex.y / (s.y + EPS_F);
  ex.z = ex.z / (s.z + EPS_F);
  ex.w = ex.w / (s.w + EPS_F);
  *(float4*)(out + (size_t)e * 4) = ex;
}

extern "C" void kernel_launch(void* const* d_in, const int* in_sizes, int n_in,
                              void* d_out, int out_size, void* d_ws, size_t ws_size,
                              hipStream_t stream) {
  const float* node_attr   = (const float*)d_in[0];
  const float* edge_weight = (const float*)d_in[1];
  const float* att_src     = (const float*)d_in[2];
  const float* att_dst     = (const float*)d_in[3];
  const float* att_edge    = (const float*)d_in[4];
  const int*   row         = (const int*)d_in[5];
  const int*   col         = (const int*)d_in[6];

  const int N = in_sizes[0] / 128;   // nodes
  const int E = in_sizes[1];         // edges
  float* out = (float*)d_out;

  // Workspace carve-up (256B aligned): sc[N*8] f32, segmax[N*4] u32, segsum[N*4] f32
  char* ws = (char*)d_ws;
  size_t off = 0;
  float* sc = (float*)(ws + off);
  off += (((size_t)N * 8 * sizeof(float)) + 255) & ~(size_t)255;
  unsigned* segmax = (unsigned*)(ws + off);
  off += (((size_t)N * 4 * sizeof(unsigned)) + 255) & ~(size_t)255;
  float* segsum = (float*)(ws + off);

  const int BLK = 256;

  // K1: init tables
  {
    int n4 = N * 4;
    init_tables<<<(n4 + BLK - 1) / BLK, BLK, 0, stream>>>(segmax, segsum, n4);
  }
  // K2: WMMA node scores — 8 waves/block, 1 tile (16 nodes) per wave
  {
    int tiles = (N + 15) / 16;
    int wavesPerBlock = BLK / 32;
    int blocks = (tiles + wavesPerBlock - 1) / wavesPerBlock;
    node_scores_wmma<<<blocks, BLK, 0, stream>>>(node_attr, att_src, att_dst, sc, N);
  }
  // K3..K5: edge passes
  {
    int blocks = (E + BLK - 1) / BLK;
    edge_alpha_max<<<blocks, BLK, 0, stream>>>(sc, edge_weight, att_edge, row, col,
                                               out, segmax, E);
    edge_exp_sum<<<blocks, BLK, 0, stream>>>(col, segmax, out, segsum, E);
    edge_norm<<<blocks, BLK, 0, stream>>>(col, segsum, out, E);
  }
}